// ForwardDistance_9371618640006
// MI455X (gfx1250) — compile-verified
//
#include <hip/hip_runtime.h>
#include <hip/hip_bf16.h>

typedef float v2f __attribute__((ext_vector_type(2)));
typedef float v8f __attribute__((ext_vector_type(8)));
typedef int   v4i __attribute__((vector_size(4 * sizeof(int))));

#ifndef __has_builtin
#define __has_builtin(x) 0
#endif

#if __has_builtin(__builtin_amdgcn_tanhf)
__device__ __forceinline__ float fast_tanh(float x) { return __builtin_amdgcn_tanhf(x); }
#else
__device__ __forceinline__ float fast_tanh(float x) { return tanhf(x); }
#endif

#define AS1 __attribute__((address_space(1)))
#define AS3 __attribute__((address_space(3)))

#if __has_builtin(__builtin_amdgcn_global_load_async_to_lds_b128)
#define HAVE_ASYNC_LDS 1
#endif

__device__ __forceinline__ void async_copy16(const void* gsrc, void* ldst) {
#if defined(HAVE_ASYNC_LDS)
  const v4i* g = (const v4i*)gsrc;   // reinterpret in generic AS
  v4i*       l = (v4i*)ldst;
  __builtin_amdgcn_global_load_async_to_lds_b128((AS1 v4i*)g, (AS3 v4i*)l, 0, 0);
#else
  *(v4i*)ldst = *(const v4i*)gsrc;
#endif
}

__device__ __forceinline__ void wait_async_then_barrier() {
#if defined(HAVE_ASYNC_LDS)
#if __has_builtin(__builtin_amdgcn_s_wait_asynccnt)
  __builtin_amdgcn_s_wait_asynccnt(0);
#else
  asm volatile("s_wait_asynccnt 0x0" ::: "memory");
#endif
#endif
  __syncthreads();
}

constexpr int Bc = 4, Nc = 512, Mc = 512, Kc = 512, Ac = 256;

// ---------------------------------------------------------------------------
// Projection: Y[r,a] = sum_k X[r,k] * W[k,a] + bias[a]
// One wave computes a 16x16 tile of Y using V_WMMA_F32_16X16X4_F32 (K-step 4).
// (Validated: emits v_wmma_f32_16x16x4_f32 x8 per unrolled iteration.)
// ---------------------------------------------------------------------------
__global__ __launch_bounds__(128) void proj_wmma(const float* __restrict__ X,
                                                 const float* __restrict__ W,
                                                 const float* __restrict__ bias,
                                                 float* __restrict__ Y) {
  const int lane = threadIdx.x;                    // 0..31 (wave32)
  const int row0 = blockIdx.x * 16;                // output row tile
  const int col0 = (blockIdx.y * 4 + threadIdx.y) * 16;
  const int l16  = lane & 15;
  const int kh   = (lane >> 4) << 1;               // 0 or 2 : K-pair selector
  const int colw = col0 + l16;                     // this lane's B/C column
  const float* xrow = X + (size_t)(row0 + l16) * Kc + kh;
  v8f acc = {0.f, 0.f, 0.f, 0.f, 0.f, 0.f, 0.f, 0.f};
#if __has_builtin(__builtin_amdgcn_wmma_f32_16x16x4_f32)
  for (int k = 0; k < Kc; k += 4) {
    v2f a = *(const v2f*)(xrow + k);               // A[m, k+kh], A[m, k+kh+1]
    v2f b;
    b.x = W[(size_t)(k + kh) * Ac + colw];         // B[k+kh,   n]
    b.y = W[(size_t)(k + kh + 1) * Ac + colw];     // B[k+kh+1, n]
    acc = __builtin_amdgcn_wmma_f32_16x16x4_f32(false, a, false, b,
                                                (short)0, acc, false, false);
  }
#else
  (void)xrow;
  {
    const int rb = row0 + ((lane >> 4) << 3);
    for (int r = 0; r < 8; ++r) {
      float s = 0.f;
      for (int k = 0; k < Kc; ++k)
        s += X[(size_t)(rb + r) * Kc + k] * W[(size_t)k * Ac + colw];
      acc[r] = s;
    }
  }
#endif
  const float bv = bias[colw];
  const int rbase = row0 + ((lane >> 4) << 3);     // lanes 16..31 hold M = 8..15
  #pragma unroll
  for (int r = 0; r < 8; ++r)
    Y[(size_t)(rbase + r) * Ac + colw] = acc[r] + bv;
}

// ---------------------------------------------------------------------------
// Fused broadcast-add -> tanh -> weighted reduce over A.
// Block: 256 threads, 64(n) x 64(m) output tile, A chunked by 64.
// Tile staging uses GLOBAL_LOAD_ASYNC_TO_LDS_B128 (ASYNCcnt) when available,
// keeping the VALU free for the 64-tanh-per-8-LDS-read inner loop.
// ---------------------------------------------------------------------------
#define SROW 68   // 64 + 4 pad: row stride == 4 mod 64 banks -> conflict-free b128

__global__ __launch_bounds__(256) void fused_tanh_dot(const float* __restrict__ dl,
                                                      const float* __restrict__ cl,
                                                      const float* __restrict__ agg,
                                                      float* __restrict__ out) {
  __shared__ float sD[64][SROW];
  __shared__ float sC[64][SROW];
  __shared__ float sG[Ac];
  const int tid = threadIdx.x;
  const int b  = blockIdx.z;
  const int n0 = blockIdx.y * 64;
  const int m0 = blockIdx.x * 64;
  const float* dlp = dl + ((size_t)b * Nc + n0) * Ac;   // 64 rows x Ac
  const float* clp = cl + ((size_t)b * Mc + m0) * Ac;
  for (int i = tid; i < Ac; i += 256) sG[i] = agg[i];
  const int tx = tid & 15, ty = tid >> 4;
  float acc[4][4] = {};
  for (int c0 = 0; c0 < Ac; c0 += 64) {
    __syncthreads();                                     // protect previous chunk
    for (int v = tid; v < 64 * 16; v += 256) {           // 64x64 f32 chunk, b128/lane
      const int r = v >> 4, q = (v & 15) << 2;
      async_copy16(&dlp[(size_t)r * Ac + c0 + q], &sD[r][q]);
      async_copy16(&clp[(size_t)r * Ac + c0 + q], &sC[r][q]);
    }
    wait_async_then_barrier();
    for (int a = 0; a < 64; a += 4) {
      const float4 g = *(const float4*)&sG[c0 + a];
      float4 dv[4], cv[4];
      #pragma unroll
      for (int i = 0; i < 4; ++i) {
        dv[i] = *(const float4*)&sD[ty + 16 * i][a];
        cv[i] = *(const float4*)&sC[tx + 16 * i][a];
      }
      #pragma unroll
      for (int i = 0; i < 4; ++i) {
        #pragma unroll
        for (int j = 0; j < 4; ++j) {
          float s;
          s  = g.x * fast_tanh(dv[i].x + cv[j].x);
          s += g.y * fast_tanh(dv[i].y + cv[j].y);
          s += g.z * fast_tanh(dv[i].z + cv[j].z);
          s += g.w * fast_tanh(dv[i].w + cv[j].w);
          acc[i][j] += s;
        }
      }
    }
  }
  #pragma unroll
  for (int i = 0; i < 4; ++i) {
    const size_t rowoff = ((size_t)b * Nc + (size_t)(n0 + ty + 16 * i)) * Mc
                        + (size_t)(m0 + tx);
    #pragma unroll
    for (int j = 0; j < 4; ++j)
      out[rowoff + 16 * j] = acc[i][j];
  }
}

// ---------------------------------------------------------------------------
extern "C" void kernel_launch(void* const* d_in, const int* in_sizes, int n_in,
                              void* d_out, int out_size, void* d_ws, size_t ws_size,
                              hipStream_t stream) {
  (void)in_sizes; (void)n_in; (void)out_size; (void)ws_size;
  const float* data = (const float*)d_in[0];   // [B,N,L]
  const float* crit = (const float*)d_in[1];   // [B,M,R]
  const float* Wl   = (const float*)d_in[2];   // [L,A]
  const float* bl   = (const float*)d_in[3];   // [A]
  const float* Wr   = (const float*)d_in[4];   // [R,A]
  const float* br   = (const float*)d_in[5];   // [A]
  const float* agg  = (const float*)d_in[6];   // [A]
  float* out = (float*)d_out;                  // [B,N,M]

  float* dl = (float*)d_ws;                    // [B,N,A]  (2 MB)
  float* cl = dl + (size_t)Bc * Nc * Ac;       // [B,M,A]  (2 MB)

  dim3 pb(32, 4, 1);
  dim3 pg(Bc * Nc / 16, Ac / 64, 1);           // (128, 4)
  proj_wmma<<<pg, pb, 0, stream>>>(data, Wl, bl, dl);
  proj_wmma<<<pg, pb, 0, stream>>>(crit, Wr, br, cl);

  dim3 fg(Mc / 64, Nc / 64, Bc);               // (8, 8, 4)
  fused_tanh_dot<<<fg, 256, 0, stream>>>(dl, cl, agg, out);
}